// Single3DRoIPointExtractor_41721312314221
// MI455X (gfx1250) — compile-verified
//
#include <hip/hip_runtime.h>

// ---------------------------------------------------------------------------
// Single3DRoIPointExtractor for MI455X (gfx1250)
//   B=4, N=32768, C=128, M=128 rois/batch, S=512 samples, out row = 3+128=131
//
// Kernel 1 (scan): WMMA-accelerated in-box test (V_WMMA_F32_16X16X4_F32,
//   16 points x 16 rois per tile, 3 wmmas: local_x / local_y / z-zc) with
//   coordinates staged into LDS via GLOBAL_LOAD_ASYNC_TO_LDS_B32 double
//   buffering (ASYNCcnt). Stable per-roi compaction of the first <=512
//   in-box point indices via a per-wave count pass + LDS prefix + write pass.
// Kernel 2 (gather): one wave per sample row; float4 feats gather, rotate xy.
// ---------------------------------------------------------------------------

typedef __attribute__((ext_vector_type(2))) float v2f;
typedef __attribute__((ext_vector_type(8))) float v8f;

typedef __attribute__((address_space(1))) int gint_t;  // global int
typedef __attribute__((address_space(3))) int lint_t;  // LDS int

#define NPTS   32768            // points per batch
#define NCH    128              // feature channels
#define NSAMP  512              // sampled points per roi
#define OUTC   (3 + NCH)        // 131 floats per output row
#define RPB    16               // rois per scan block (one WMMA tile of columns)
#define WAVES  8                // waves per block (256 threads, wave32)
#define PTSW   (NPTS / WAVES)   // 4096 points per wave
#define CHPTS  256              // points per LDS chunk per wave
#define CHF    (CHPTS * 3)      // floats per chunk (768 = 3 KB)
#define NCHUNK (PTSW / CHPTS)   // 16 chunks per wave
#define TILESC (CHPTS / 16)     // 16-point WMMA tiles per chunk

// Async-stage one chunk (CHF floats) from global to this wave's LDS buffer.
// 12-byte point stride => only 4B global alignment guaranteed => b32 ops.
__device__ __forceinline__ void stage_chunk(const float* __restrict__ g,
                                            float* l, int lane) {
#pragma unroll
  for (int i = 0; i < CHF / 32; ++i) {
    const int off = i * 32 + lane;
    __builtin_amdgcn_global_load_async_to_lds_b32(
        (gint_t*)(g + off),
        (lint_t*)(l + off),
        /*offset=*/0, /*cpol=*/0);
  }
}

// One full sweep over this wave's point range.
//  WRITE=false: returns in-box count (valid in lanes 0..15, per-roi).
//  WRITE=true : writes indices starting at 'pos', capped at NSAMP entries.
template <bool WRITE>
__device__ __forceinline__ int scan_pass(
    const float* __restrict__ gsrc,  // coord base of this wave's range
    float* l0, float* l1,            // this wave's two LDS chunk buffers
    int lane, v2f bx, v2f by, v2f bz,
    float hdx, float hdy, float hdz,
    int p0base,                      // batch-local point index of range start
    int pos, int* __restrict__ idxOut) {
  const int n = lane & 15;
  stage_chunk(gsrc, l0, lane);                       // prologue: chunk 0
  for (int c = 0; c < NCHUNK; ++c) {
    float* cur = (c & 1) ? l1 : l0;
    if (c + 1 < NCHUNK) {
      stage_chunk(gsrc + (size_t)(c + 1) * CHF, (c & 1) ? l0 : l1, lane);
      // chunk c's CHF/32 async loads complete in order before the new ones
      __builtin_amdgcn_s_wait_asynccnt(CHF / 32);
    } else {
      __builtin_amdgcn_s_wait_asynccnt(0);
    }
#pragma unroll 1
    for (int t = 0; t < TILESC; ++t) {
      // A (16x4 f32): lane m (0..15) -> {x_m, y_m}; lane m+16 -> {z_m, 1}
      const float* pc = cur + (t * 16 + n) * 3;
      v2f a;
      if (lane < 16) a = (v2f){pc[0], pc[1]};
      else           a = (v2f){pc[2], 1.0f};
      v8f acc = {};
      v8f lx = __builtin_amdgcn_wmma_f32_16x16x4_f32(false, a, false, bx,
                                                     (short)0, acc, false, false);
      v8f ly = __builtin_amdgcn_wmma_f32_16x16x4_f32(false, a, false, by,
                                                     (short)0, acc, false, false);
      v8f lz = __builtin_amdgcn_wmma_f32_16x16x4_f32(false, a, false, bz,
                                                     (short)0, acc, false, false);
      // D layout: lane = roi column (n = lane&15), VGPR r = row m = r + 8*(lane>>4)
      unsigned m8 = 0;
#pragma unroll
      for (int r = 0; r < 8; ++r) {
        const int in = (__builtin_fabsf(lx[r]) < hdx) &
                       (__builtin_fabsf(ly[r]) < hdy) &
                       (__builtin_fabsf(lz[r]) <= hdz);
        m8 |= ((unsigned)in) << r;
      }
      // fuse rows 0..7 (lane n) and rows 8..15 (lane n+16) into one 16-bit mask
      const unsigned lo = (unsigned)__builtin_amdgcn_ds_bpermute(n << 2, (int)m8);
      const unsigned hi = (unsigned)__builtin_amdgcn_ds_bpermute((n + 16) << 2, (int)m8);
      unsigned m16 = (lo & 0xffu) | ((hi & 0xffu) << 8);
      if (lane < 16) {
        if (!WRITE) {
          pos += __builtin_popcount(m16);
        } else {
          const int pglob = p0base + c * CHPTS + t * 16;
          while (m16) {
            const int r = __builtin_ctz(m16);
            m16 &= m16 - 1;
            if (pos < NSAMP) idxOut[pos] = pglob + r;  // stable order
            ++pos;
          }
        }
      }
    }
  }
  return pos;
}

__global__ void __launch_bounds__(256)
roi_scan_kernel(const float* __restrict__ coord,   // (B, NPTS, 3)
                const float* __restrict__ rois,    // (R, 8)
                int* __restrict__ d_idx,           // (R, NSAMP)
                int* __restrict__ d_cnt) {         // (R)
  __shared__ float lchunk[WAVES][2][CHF];          // 48 KB staging
  __shared__ int lcnt[WAVES * RPB];
  __shared__ int lbase[WAVES * RPB];

  const int tid = threadIdx.x;
  const int lane = tid & 31;
  const int w = tid >> 5;
  const int n = lane & 15;
  const int roi = blockIdx.x * RPB + n;

  // All 16 rois of a block share a batch (M=128 rois/batch, 16 | 128).
  const int batch = (int)rois[(size_t)blockIdx.x * RPB * 8];

  // Per-lane roi params (lane l and l+16 hold roi l&15's constants).
  const float* rp = rois + (size_t)roi * 8;
  const float cx = rp[1], cy = rp[2], cz = rp[3];
  const float dx = rp[4], dy = rp[5], dz = rp[6], rz = rp[7];
  const float ca = cosf(rz);          // cos(-rz)
  const float sa = -sinf(rz);         // sin(-rz)
  const float tx = -cx * ca + cy * sa;
  const float ty = -cx * sa - cy * ca;
  const float zc = cz + 0.5f * dz;    // geometric center z
  const float hdx = 0.5f * dx, hdy = 0.5f * dy, hdz = 0.5f * dz;

  // B (4x16 f32) columns: lanes 0..15 hold rows K0,K1; lanes 16..31 rows K2,K3
  //   local_x = [x y z 1] . [ca, -sa, 0, tx]
  //   local_y = [x y z 1] . [sa,  ca, 0, ty]
  //   z-zc    = [x y z 1] . [ 0,   0, 1, -zc]
  v2f bx, by, bz;
  if (lane < 16) { bx = (v2f){ca, -sa}; by = (v2f){sa, ca}; bz = (v2f){0.f, 0.f}; }
  else           { bx = (v2f){0.f, tx}; by = (v2f){0.f, ty}; bz = (v2f){1.f, -zc}; }

  const float* gbase = coord + ((size_t)batch * NPTS + (size_t)w * PTSW) * 3;
  float* myl0 = &lchunk[w][0][0];
  float* myl1 = &lchunk[w][1][0];

  // Pass 1: per-wave, per-roi in-box counts.
  const int cnt = scan_pass<false>(gbase, myl0, myl1, lane, bx, by, bz,
                                   hdx, hdy, hdz, w * PTSW, 0, nullptr);
  if (lane < 16) lcnt[w * RPB + n] = cnt;
  __syncthreads();

  // Prefix over waves (point ranges are in increasing order => stable).
  if (tid < RPB) {
    int base = 0;
    for (int ww = 0; ww < WAVES; ++ww) {
      lbase[ww * RPB + tid] = base;
      base += lcnt[ww * RPB + tid];
    }
    d_cnt[blockIdx.x * RPB + tid] = base < NSAMP ? base : NSAMP;
  }
  __syncthreads();

  // Pass 2: write the first <=NSAMP in-box indices, in original point order.
  const int start = lbase[w * RPB + n];
  scan_pass<true>(gbase, myl0, myl1, lane, bx, by, bz, hdx, hdy, hdz,
                  w * PTSW, start, d_idx + (size_t)roi * NSAMP);
}

__global__ void __launch_bounds__(256)
roi_gather_kernel(const float* __restrict__ coord,  // (B, NPTS, 3)
                  const float* __restrict__ feats,  // (B, NPTS, NCH)
                  const float* __restrict__ rois,   // (R, 8)
                  const int* __restrict__ d_idx,    // (R, NSAMP)
                  const int* __restrict__ d_cnt,    // (R)
                  float* __restrict__ out) {        // (R, NSAMP, OUTC)
  const int r = blockIdx.x;
  const int tid = threadIdx.x;
  const int cnt = d_cnt[r];
  float* outR = out + (size_t)r * NSAMP * OUTC;

  if (cnt == 0) {  // empty roi -> zero the whole row block (d_out is poisoned)
    for (int i = tid; i < NSAMP * OUTC; i += 256) outR[i] = 0.0f;
    return;
  }

  const float* rp = rois + (size_t)r * 8;
  const int batch = (int)rp[0];
  const float cx = rp[1], cy = rp[2], cz = rp[3], rz = rp[7];
  const float c2 = cosf(rz);    // cos(-theta)
  const float s2 = -sinf(rz);   // sin(-theta)

  const int lane = tid & 31;
  const int w = tid >> 5;
  const float* cb = coord + (size_t)batch * NPTS * 3;
  const float* fb = feats + (size_t)batch * NPTS * NCH;
  const int* il = d_idx + (size_t)r * NSAMP;

  for (int s = w; s < NSAMP; s += WAVES) {
    const int take = s % cnt;            // cyclic duplication, cnt in [1,512]
    const int pi = il[take];             // wave-uniform
    const float px = cb[pi * 3 + 0];
    const float py = cb[pi * 3 + 1];
    const float pz = cb[pi * 3 + 2];
    const float ux = px - cx, uy = py - cy;
    float* orow = outR + (size_t)s * OUTC;
    if (lane < 3) {
      const float v = (lane == 0) ? (ux * c2 - uy * s2)
                    : (lane == 1) ? (ux * s2 + uy * c2)
                                  : (pz - cz);           // cz = bottom center
      orow[lane] = v;
    }
    // 128 feats: one aligned float4 per lane (rows are 512B strided)
    const float4 f = reinterpret_cast<const float4*>(fb + (size_t)pi * NCH)[lane];
    float* od = orow + 3 + lane * 4;
    od[0] = f.x; od[1] = f.y; od[2] = f.z; od[3] = f.w;
  }
}

extern "C" void kernel_launch(void* const* d_in, const int* in_sizes, int n_in,
                              void* d_out, int out_size, void* d_ws, size_t ws_size,
                              hipStream_t stream) {
  const float* feats = (const float*)d_in[0];   // (B, N, C) f32
  const float* coord = (const float*)d_in[1];   // (B, N, 3) f32
  const float* rois  = (const float*)d_in[3];   // (R, 8)    f32
  const int R = in_sizes[3] / 8;                // 512 total rois

  int* d_idx = (int*)d_ws;                      // R * NSAMP ints
  int* d_cnt = d_idx + (size_t)R * NSAMP;       // R ints

  roi_scan_kernel<<<R / RPB, 256, 0, stream>>>(coord, rois, d_idx, d_cnt);
  roi_gather_kernel<<<R, 256, 0, stream>>>(coord, feats, rois, d_idx, d_cnt,
                                           (float*)d_out);
}